// PatientAdaptiveGNN_25340307047148
// MI455X (gfx1250) — compile-verified
//
#include <hip/hip_runtime.h>

#define N_NODES 50000
#define DMODEL  256
#define RLORA   16
#define NLAYERS 2
#define NEDGES  300000
#define NPAIRS  100000

typedef __attribute__((ext_vector_type(2))) float v2f;
typedef __attribute__((ext_vector_type(8))) float v8f;

// One wave computes a 16 x (16*CT) strip of Y[t] = X[xt] * W[t]^T (+bias)(+add0).
// A fragment (16x4 f32): lanes 0-15 hold K={k,k+1}, lanes 16-31 hold K={k+2,k+3}.
// B fragment (4x16 f32): lane (l16,half) loads W[o, k+2*half .. +1] (float2).
// The A fragment is reused across CT column tiles -> 1/CT the activation traffic.
template <int CT>
__global__ __launch_bounds__(32) void gemm_wmma_f32(
    const float* __restrict__ X, const float* __restrict__ W,
    const float* __restrict__ bias, const float* __restrict__ add0,
    float* __restrict__ Y, int nrows, int K, int ocols, int flipXT)
{
  const int lane = threadIdx.x;
  const int m0 = blockIdx.x << 4;
  const int o0 = blockIdx.y * (16 * CT);
  const int t  = blockIdx.z;
  const int xt = flipXT ? (1 - t) : t;
  const int half = lane >> 4;
  const int l16  = lane & 15;

  const float* xrow = X + (size_t)xt * nrows * K + (size_t)(m0 + l16) * K + 2 * half;
  const float* wrow = W + (size_t)t  * ocols * K + (size_t)(o0 + l16) * K + 2 * half;

  v8f c[CT];
#pragma unroll
  for (int ct = 0; ct < CT; ++ct) c[ct] = (v8f){};

  for (int k = 0; k < K; k += 4) {
    v2f a = *(const v2f*)(xrow + k);
#pragma unroll
    for (int ct = 0; ct < CT; ++ct) {
      v2f b = *(const v2f*)(wrow + (size_t)ct * 16 * K + k);
      c[ct] = __builtin_amdgcn_wmma_f32_16x16x4_f32(
          /*neg_a=*/false, a, /*neg_b=*/false, b,
          /*c_mod=*/(short)0, c[ct], /*reuse_a=*/false, /*reuse_b=*/false);
    }
  }

  // C/D layout: VGPR r, lanes 0-15 -> M=r, lanes 16-31 -> M=r+8; N = l16.
  const size_t rowbase = (size_t)t * nrows * ocols + (size_t)(m0 + 8 * half) * ocols;
#pragma unroll
  for (int ct = 0; ct < CT; ++ct) {
    const int oc = o0 + ct * 16 + l16;
    const float bb = bias ? bias[(size_t)t * ocols + oc] : 0.0f;
#pragma unroll
    for (int r = 0; r < 8; ++r) {
      size_t idx = rowbase + (size_t)r * ocols + oc;
      float v = c[ct][r] + bb;
      if (add0) v += add0[idx];
      Y[idx] = v;
    }
  }
}

// Scatter-add messages into agg + per-destination edge counts.
// 64 threads per edge, each moving one float4 chunk of the 256-wide row.
__global__ __launch_bounds__(256) void scatter_mean_acc(
    const float* __restrict__ h, const int* __restrict__ e_src,
    const int* __restrict__ e_dst, float* __restrict__ agg,
    float* __restrict__ cnt)
{
  const int t = blockIdx.y;
  size_t tid = (size_t)blockIdx.x * 256 + threadIdx.x;
  size_t e = tid >> 6;
  int chunk = (int)(tid & 63);
  if (e >= NEDGES) return;
  int s = e_src[(size_t)t * NEDGES + e];
  int d = e_dst[(size_t)t * NEDGES + e];
  const float4 v = *(const float4*)(h + ((size_t)t * N_NODES + s) * DMODEL + chunk * 4);
  float* o = agg + ((size_t)t * N_NODES + d) * DMODEL + chunk * 4;
  atomicAdd(o + 0, v.x);
  atomicAdd(o + 1, v.y);
  atomicAdd(o + 2, v.z);
  atomicAdd(o + 3, v.w);
  if (chunk == 0) atomicAdd(cnt + (size_t)t * N_NODES + d, 1.0f);
}

__global__ __launch_bounds__(256) void div_count(float* __restrict__ agg,
                                                 const float* __restrict__ cnt)
{
  size_t i = (size_t)blockIdx.x * 256 + threadIdx.x;   // float4 chunks
  const size_t total = (size_t)2 * N_NODES * (DMODEL / 4);
  if (i >= total) return;
  float c = fmaxf(cnt[i >> 6], 1.0f);
  float4* p = (float4*)agg + i;
  float4 v = *p;
  v.x /= c; v.y /= c; v.z /= c; v.w /= c;
  *p = v;
}

// LayerNorm over D=256, one wave32 per row (8 elems/lane), optional residual.
__global__ __launch_bounds__(128) void ln_rows(
    const float* __restrict__ in, const float* __restrict__ res,
    const float* __restrict__ g, const float* __restrict__ bta,
    float* __restrict__ out)
{
  const int lane = threadIdx.x & 31;
  const int wv = threadIdx.x >> 5;
  size_t row = (size_t)blockIdx.x * 4 + wv;
  if (row >= (size_t)2 * N_NODES) return;
  const int t = (row >= (size_t)N_NODES) ? 1 : 0;
  const float* x = in + row * DMODEL + lane * 8;
  float v[8];
  float4 q0 = *(const float4*)(x);
  float4 q1 = *(const float4*)(x + 4);
  v[0] = q0.x; v[1] = q0.y; v[2] = q0.z; v[3] = q0.w;
  v[4] = q1.x; v[5] = q1.y; v[6] = q1.z; v[7] = q1.w;
  if (res) {
    const float* r = res + row * DMODEL + lane * 8;
    float4 r0 = *(const float4*)(r);
    float4 r1 = *(const float4*)(r + 4);
    v[0] += r0.x; v[1] += r0.y; v[2] += r0.z; v[3] += r0.w;
    v[4] += r1.x; v[5] += r1.y; v[6] += r1.z; v[7] += r1.w;
  }
  float s = 0.0f;
#pragma unroll
  for (int j = 0; j < 8; ++j) s += v[j];
#pragma unroll
  for (int off = 16; off > 0; off >>= 1) s += __shfl_xor(s, off, 32);
  const float mu = s * (1.0f / DMODEL);
  float ss = 0.0f;
#pragma unroll
  for (int j = 0; j < 8; ++j) { float d = v[j] - mu; ss += d * d; }
#pragma unroll
  for (int off = 16; off > 0; off >>= 1) ss += __shfl_xor(ss, off, 32);
  const float inv = rsqrtf(ss * (1.0f / DMODEL) + 1e-5f);
  const float* gp = g + (size_t)t * DMODEL + lane * 8;
  const float* bp = bta + (size_t)t * DMODEL + lane * 8;
  float* op = out + row * DMODEL + lane * 8;
#pragma unroll
  for (int j = 0; j < 8; ++j) op[j] = (v[j] - mu) * inv * gp[j] + bp[j];
}

// score[k] = dot(h_a[src[k]], h_b[dst[k]]), one wave32 per pair.
__global__ __launch_bounds__(128) void score_pairs(
    const float* __restrict__ h, const int* __restrict__ src,
    const int* __restrict__ dst, float* __restrict__ out)
{
  const int lane = threadIdx.x & 31;
  const int wv = threadIdx.x >> 5;
  size_t k = (size_t)blockIdx.x * 4 + wv;
  if (k >= NPAIRS) return;
  const float* pa = h + (size_t)src[k] * DMODEL + lane * 8;                 // tower 0
  const float* pb = h + ((size_t)N_NODES + dst[k]) * DMODEL + lane * 8;     // tower 1
  float4 a0 = *(const float4*)(pa);
  float4 a1 = *(const float4*)(pa + 4);
  float4 b0 = *(const float4*)(pb);
  float4 b1 = *(const float4*)(pb + 4);
  float acc = a0.x * b0.x + a0.y * b0.y + a0.z * b0.z + a0.w * b0.w
            + a1.x * b1.x + a1.y * b1.y + a1.z * b1.z + a1.w * b1.w;
#pragma unroll
  for (int off = 16; off > 0; off >>= 1) acc += __shfl_xor(acc, off, 32);
  if (lane == 0) out[k] = acc;
}

extern "C" void kernel_launch(void* const* d_in, const int* in_sizes, int n_in,
                              void* d_out, int out_size, void* d_ws, size_t ws_size,
                              hipStream_t stream)
{
  (void)in_sizes; (void)n_in; (void)out_size; (void)ws_size;
  const float* x      = (const float*)d_in[0];
  const float* W_in   = (const float*)d_in[1];
  const float* b_in   = (const float*)d_in[2];
  const float* lora_A = (const float*)d_in[3];
  const float* lora_B = (const float*)d_in[4];
  const float* pn_g   = (const float*)d_in[5];
  const float* pn_b   = (const float*)d_in[6];
  const float* W_self = (const float*)d_in[7];
  const float* W_neigh= (const float*)d_in[8];
  const float* b_sage = (const float*)d_in[9];
  const float* ln_g   = (const float*)d_in[10];
  const float* ln_b   = (const float*)d_in[11];
  const float* Wv     = (const float*)d_in[12];
  const float* bv     = (const float*)d_in[13];
  const float* Wo     = (const float*)d_in[14];
  const float* bo     = (const float*)d_in[15];
  const int* e_src    = (const int*)d_in[16];
  const int* e_dst    = (const int*)d_in[17];
  const int* pos_src  = (const int*)d_in[18];
  const int* pos_dst  = (const int*)d_in[19];
  const int* neg_src  = (const int*)d_in[20];
  const int* neg_dst  = (const int*)d_in[21];
  float* out = (float*)d_out;

  const size_t SZ = (size_t)2 * N_NODES * DMODEL;
  float* ws  = (float*)d_ws;
  float* h   = ws;            // persistent hidden state
  float* b1  = h + SZ;        // h_new accumulator
  float* b2  = b1 + SZ;       // agg (reused as v in last layer)
  float* b3  = b2 + SZ;       // pre-LN temp
  float* xr  = b3 + SZ;       // LoRA intermediate (2,N,R)
  float* cnt = xr + (size_t)2 * N_NODES * RLORA;

  const dim3 blk32(32);
  const int MT = N_NODES / 16;          // 3125 row tiles
  const dim3 gFull(MT, DMODEL / 128, 2); // CT=8 -> 2 column strips of 128

  // LoRA: xr = x @ A^T (O=16 -> CT=1) ; b1 = xr @ B^T (K=16, CT=8)
  gemm_wmma_f32<1><<<dim3(MT, 1, 2), blk32, 0, stream>>>(
      x, lora_A, nullptr, nullptr, xr, N_NODES, DMODEL, RLORA, 0);
  gemm_wmma_f32<8><<<gFull, blk32, 0, stream>>>(
      xr, lora_B, nullptr, nullptr, b1, N_NODES, RLORA, DMODEL, 0);
  // base + lora: b3 = x @ W_in^T + b_in + b1 ; h = LN(b3)
  gemm_wmma_f32<8><<<gFull, blk32, 0, stream>>>(
      x, W_in, b_in, b1, b3, N_NODES, DMODEL, DMODEL, 0);
  ln_rows<<<dim3(2 * N_NODES / 4), dim3(128), 0, stream>>>(b3, nullptr, pn_g, pn_b, h);

  for (int i = 0; i < NLAYERS; ++i) {
    hipMemsetAsync(b2, 0, SZ * sizeof(float), stream);
    hipMemsetAsync(cnt, 0, (size_t)2 * N_NODES * sizeof(float), stream);
    scatter_mean_acc<<<dim3((NEDGES * 64 + 255) / 256, 2), dim3(256), 0, stream>>>(
        h, e_src, e_dst, b2, cnt);
    div_count<<<dim3((2 * N_NODES * (DMODEL / 4) + 255) / 256), dim3(256), 0, stream>>>(b2, cnt);

    const float* Ws = W_self + (size_t)i * 2 * DMODEL * DMODEL;
    const float* Wn = W_neigh + (size_t)i * 2 * DMODEL * DMODEL;
    const float* bs = b_sage + (size_t)i * 2 * DMODEL;
    // b1 = h @ W_self^T + b ; b1 += neigh(agg tower-swapped) @ W_neigh^T
    gemm_wmma_f32<8><<<gFull, blk32, 0, stream>>>(
        h, Ws, bs, nullptr, b1, N_NODES, DMODEL, DMODEL, 0);
    gemm_wmma_f32<8><<<gFull, blk32, 0, stream>>>(
        b2, Wn, nullptr, b1, b1, N_NODES, DMODEL, DMODEL, 1);

    const float* pre = b1;
    if (i == NLAYERS - 1) {
      gemm_wmma_f32<8><<<gFull, blk32, 0, stream>>>(
          b1, Wv, bv, nullptr, b2, N_NODES, DMODEL, DMODEL, 0);
      gemm_wmma_f32<8><<<gFull, blk32, 0, stream>>>(
          b2, Wo, bo, nullptr, b3, N_NODES, DMODEL, DMODEL, 0);
      pre = b3;
    }
    // h = LN(pre (+ h residual if i>0))
    ln_rows<<<dim3(2 * N_NODES / 4), dim3(128), 0, stream>>>(
        pre, (i > 0) ? h : nullptr,
        ln_g + (size_t)i * 2 * DMODEL, ln_b + (size_t)i * 2 * DMODEL, h);
  }

  score_pairs<<<dim3(NPAIRS / 4), dim3(128), 0, stream>>>(h, pos_src, pos_dst, out);
  score_pairs<<<dim3(NPAIRS / 4), dim3(128), 0, stream>>>(h, neg_src, neg_dst, out + NPAIRS);
}